// ParallelConvMambaBlock_10462540333190
// MI455X (gfx1250) — compile-verified
//
#include <hip/hip_runtime.h>
#include <hip/hip_bf16.h>
#include <math.h>

typedef __attribute__((ext_vector_type(2))) float v2f;
typedef __attribute__((ext_vector_type(8))) float v8f;

#define Bsz   2
#define Hsp   56
#define Wsp   56
#define Lsz   3136
#define DIMc  96
#define DINN  192
#define KDIR  4
#define DST   16
#define DTR   6
#define XDW   48      // x_dbl padded width (38 -> 48 for 3 N-tiles)
#define NROW  6272    // B*L

// ---------------- WMMA helper (fp32 16x16x4) ----------------
__device__ __forceinline__ v8f wmma4(v2f a, v2f b, v8f c) {
  // 8 args: (neg_a, A, neg_b, B, c_mod, C, reuse_a, reuse_b)
  return __builtin_amdgcn_wmma_f32_16x16x4_f32(false, a, false, b, (short)0, c,
                                               false, false);
}

__device__ __forceinline__ void store_tile(float* __restrict__ base, int ldc,
                                           int tm, int tn, int mn, int g, v8f acc) {
  const float* af = (const float*)&acc;
#pragma unroll
  for (int r = 0; r < 8; ++r)
    base[(size_t)(tm * 16 + r + 8 * g) * ldc + tn * 16 + mn] = af[r];
}

// scan-order index -> spatial index (identical for read of u and write of y)
__device__ __forceinline__ int scan_pos(int k, int l) {
  if (k == 0) return l;
  if (k == 2) return Lsz - 1 - l;
  int t = (k == 1) ? l : (Lsz - 1 - l);
  return (t % Hsp) * Wsp + (t / Hsp);
}

// ---------------- elementwise / LN kernels ----------------
__global__ void k_ln(const float* __restrict__ in, const float* __restrict__ g,
                     const float* __restrict__ b, float* __restrict__ out,
                     int rows, int width) {
  int r = blockIdx.x * blockDim.x + threadIdx.x;
  if (r >= rows) return;
  const float* p = in + (size_t)r * width;
  float m = 0.f;
  for (int i = 0; i < width; ++i) m += p[i];
  m /= (float)width;
  float v = 0.f;
  for (int i = 0; i < width; ++i) { float d = p[i] - m; v += d * d; }
  v /= (float)width;
  float inv = rsqrtf(v + 1e-5f);
  float* o = out + (size_t)r * width;
  for (int i = 0; i < width; ++i) o[i] = (p[i] - m) * inv * g[i] + b[i];
}

__global__ void k_zero(float* __restrict__ p, int n) {
  int i = blockIdx.x * blockDim.x + threadIdx.x;
  if (i < n) p[i] = 0.f;
}

// LN(y,192) * silu(z)
__global__ void k_gate(const float* __restrict__ Yb, const float* __restrict__ XZ,
                       const float* __restrict__ g, const float* __restrict__ b,
                       float* __restrict__ out) {
  int r = blockIdx.x * blockDim.x + threadIdx.x;
  if (r >= NROW) return;
  const float* p = Yb + (size_t)r * DINN;
  float m = 0.f;
  for (int i = 0; i < DINN; ++i) m += p[i];
  m /= (float)DINN;
  float v = 0.f;
  for (int i = 0; i < DINN; ++i) { float d = p[i] - m; v += d * d; }
  v /= (float)DINN;
  float inv = rsqrtf(v + 1e-5f);
  const float* z = XZ + (size_t)r * (2 * DINN) + DINN;
  float* o = out + (size_t)r * DINN;
  for (int i = 0; i < DINN; ++i) {
    float ln = (p[i] - m) * inv * g[i] + b[i];
    float zz = z[i];
    o[i] = ln * (zz / (1.f + expf(-zz)));
  }
}

// out = x + LN(fused, 96)
__global__ void k_final(const float* __restrict__ X, const float* __restrict__ F,
                        const float* __restrict__ g, const float* __restrict__ b,
                        float* __restrict__ out) {
  int r = blockIdx.x * blockDim.x + threadIdx.x;
  if (r >= NROW) return;
  const float* p = F + (size_t)r * DIMc;
  float m = 0.f;
  for (int i = 0; i < DIMc; ++i) m += p[i];
  m /= (float)DIMc;
  float v = 0.f;
  for (int i = 0; i < DIMc; ++i) { float d = p[i] - m; v += d * d; }
  v /= (float)DIMc;
  float inv = rsqrtf(v + 1e-5f);
  const float* xr = X + (size_t)r * DIMc;
  float* o = out + (size_t)r * DIMc;
  for (int i = 0; i < DIMc; ++i)
    o[i] = xr[i] + (p[i] - m) * inv * g[i] + b[i];
}

// ---------------- depthwise convs ----------------
// mamba branch: xm = xz[..., 0:192] (B,H,W,192) -> silu(dwconv3) -> xconv (B,192,L)
__global__ void k_conv_mamba(const float* __restrict__ XZ, const float* __restrict__ cw,
                             const float* __restrict__ cb, float* __restrict__ Xconv) {
  int idx = blockIdx.x * blockDim.x + threadIdx.x;
  if (idx >= Bsz * DINN * Lsz) return;
  int l = idx % Lsz;
  int d = (idx / Lsz) % DINN;
  int b = idx / (DINN * Lsz);
  int h = l / Wsp, w = l % Wsp;
  float s = cb[d];
#pragma unroll
  for (int kh = 0; kh < 3; ++kh) {
    int hh = h + kh - 1;
    if (hh < 0 || hh >= Hsp) continue;
#pragma unroll
    for (int kw = 0; kw < 3; ++kw) {
      int ww = w + kw - 1;
      if (ww < 0 || ww >= Wsp) continue;
      s += XZ[((size_t)(b * Lsz + hh * Wsp + ww)) * (2 * DINN) + d] *
           cw[d * 9 + kh * 3 + kw];
    }
  }
  Xconv[idx] = s / (1.f + expf(-s));  // silu
}

// conv branch: relu(bn1(dwconv3(x))) -> cbr (rows, 96)
__global__ void k_conv_branch(const float* __restrict__ X, const float* __restrict__ dw,
                              const float* __restrict__ db, const float* __restrict__ g1,
                              const float* __restrict__ b1, const float* __restrict__ m1,
                              const float* __restrict__ v1, float* __restrict__ Cbr) {
  int idx = blockIdx.x * blockDim.x + threadIdx.x;
  if (idx >= NROW * DIMc) return;
  int ch = idx % DIMc;
  int l = (idx / DIMc) % Lsz;
  int b = idx / (DIMc * Lsz);
  int h = l / Wsp, w = l % Wsp;
  float s = db[ch];
#pragma unroll
  for (int kh = 0; kh < 3; ++kh) {
    int hh = h + kh - 1;
    if (hh < 0 || hh >= Hsp) continue;
#pragma unroll
    for (int kw = 0; kw < 3; ++kw) {
      int ww = w + kw - 1;
      if (ww < 0 || ww >= Wsp) continue;
      s += X[((size_t)(b * Lsz + hh * Wsp + ww)) * DIMc + ch] *
           dw[ch * 9 + kh * 3 + kw];
    }
  }
  s = (s - m1[ch]) * rsqrtf(v1[ch] + 1e-5f) * g1[ch] + b1[ch];
  Cbr[idx] = fmaxf(s, 0.f);
}

// ---------------- WMMA GEMMs ----------------
// xz = xn(6272x96) @ in_proj_w^T(96x384)
__global__ void k_gemm_inproj(const float* __restrict__ A, const float* __restrict__ W,
                              float* __restrict__ Out) {
  int lane = threadIdx.x, mn = lane & 15, g = lane >> 4;
  const float* ar = A + (size_t)(blockIdx.x * 16 + mn) * DIMc;
  const float* br = W + (size_t)(blockIdx.y * 16 + mn) * DIMc;
  v8f acc = {};
  for (int k0 = 0; k0 < DIMc; k0 += 4) {
    int kk = k0 + 2 * g;
    v2f a, b;
    a.x = ar[kk]; a.y = ar[kk + 1];
    b.x = br[kk]; b.y = br[kk + 1];
    acc = wmma4(a, b, acc);
  }
  store_tile(Out, 2 * DINN, blockIdx.x, blockIdx.y, mn, g, acc);
}

// x_dbl[b,k,l,0:38] = sum_d xs[b,k,d,l] * x_proj_w[k,c,d]   (N padded to 48)
__global__ void k_gemm_xproj(const float* __restrict__ Xconv,
                             const float* __restrict__ XPW,
                             float* __restrict__ Xdbl) {
  int bk = blockIdx.z;
  int b = bk >> 2, k = bk & 3;
  int lane = threadIdx.x, mn = lane & 15, g = lane >> 4;
  int lrow = blockIdx.x * 16 + mn;                  // scan index (A row)
  int pos = scan_pos(k, lrow);
  int c = blockIdx.y * 16 + mn;                     // output channel (B col)
  const float* xc = Xconv + (size_t)b * DINN * Lsz;
  int valid = (c < DTR + 2 * DST) ? 1 : 0;          // c < 38
  const float* wr = XPW + (size_t)(k * (DTR + 2 * DST) + (valid ? c : 0)) * DINN;
  float bscale = valid ? 1.f : 0.f;
  v8f acc = {};
  for (int k0 = 0; k0 < DINN; k0 += 4) {
    int kk = k0 + 2 * g;
    v2f a, bb;
    a.x = xc[(size_t)kk * Lsz + pos];
    a.y = xc[(size_t)(kk + 1) * Lsz + pos];
    bb.x = wr[kk] * bscale;
    bb.y = wr[kk + 1] * bscale;
    acc = wmma4(a, bb, acc);
  }
  float* out = Xdbl + (size_t)bk * Lsz * XDW;
  store_tile(out, XDW, blockIdx.x, blockIdx.y, mn, g, acc);
}

// mamba_out = yg(6272x192) @ out_proj_w^T(192x96)
__global__ void k_gemm_outproj(const float* __restrict__ A, const float* __restrict__ W,
                               float* __restrict__ Out) {
  int lane = threadIdx.x, mn = lane & 15, g = lane >> 4;
  const float* ar = A + (size_t)(blockIdx.x * 16 + mn) * DINN;
  const float* br = W + (size_t)(blockIdx.y * 16 + mn) * DINN;
  v8f acc = {};
  for (int k0 = 0; k0 < DINN; k0 += 4) {
    int kk = k0 + 2 * g;
    v2f a, b;
    a.x = ar[kk]; a.y = ar[kk + 1];
    b.x = br[kk]; b.y = br[kk + 1];
    acc = wmma4(a, b, acc);
  }
  store_tile(Out, DIMc, blockIdx.x, blockIdx.y, mn, g, acc);
}

// conv_out = bn2(cbr(6272x96) @ pw_w^T + pw_b)
__global__ void k_gemm_pw(const float* __restrict__ A, const float* __restrict__ W,
                          const float* __restrict__ pwb, const float* __restrict__ g2,
                          const float* __restrict__ b2, const float* __restrict__ m2,
                          const float* __restrict__ v2, float* __restrict__ Out) {
  int lane = threadIdx.x, mn = lane & 15, g = lane >> 4;
  int col = blockIdx.y * 16 + mn;
  const float* ar = A + (size_t)(blockIdx.x * 16 + mn) * DIMc;
  const float* br = W + (size_t)col * DIMc;
  v8f acc = {};
  for (int k0 = 0; k0 < DIMc; k0 += 4) {
    int kk = k0 + 2 * g;
    v2f a, b;
    a.x = ar[kk]; a.y = ar[kk + 1];
    b.x = br[kk]; b.y = br[kk + 1];
    acc = wmma4(a, b, acc);
  }
  float scale = g2[col] * rsqrtf(v2[col] + 1e-5f);
  float shift = (pwb[col] - m2[col]) * scale + b2[col];
  float* af = (float*)&acc;
#pragma unroll
  for (int r = 0; r < 8; ++r) af[r] = af[r] * scale + shift;
  store_tile(Out, DIMc, blockIdx.x, blockIdx.y, mn, g, acc);
}

// fused = concat(conv_out, mamba_out)(6272x192) @ fus_w^T + fus_b
__global__ void k_gemm_fus(const float* __restrict__ C1, const float* __restrict__ C2,
                           const float* __restrict__ W, const float* __restrict__ fb,
                           float* __restrict__ Out) {
  int lane = threadIdx.x, mn = lane & 15, g = lane >> 4;
  int row = blockIdx.x * 16 + mn;
  int col = blockIdx.y * 16 + mn;
  const float* a1 = C1 + (size_t)row * DIMc;
  const float* a2 = C2 + (size_t)row * DIMc;
  const float* br = W + (size_t)col * DINN;
  v8f acc = {};
  for (int k0 = 0; k0 < DINN; k0 += 4) {
    int kk = k0 + 2 * g;
    v2f a, b;
    if (kk < DIMc) { a.x = a1[kk]; a.y = a1[kk + 1]; }
    else           { a.x = a2[kk - DIMc]; a.y = a2[kk - DIMc + 1]; }
    b.x = br[kk]; b.y = br[kk + 1];
    acc = wmma4(a, b, acc);
  }
  float bias = fb[col];
  float* af = (float*)&acc;
#pragma unroll
  for (int r = 0; r < 8; ++r) af[r] += bias;
  store_tile(Out, DIMc, blockIdx.x, blockIdx.y, mn, g, acc);
}

// ---------------- selective scan ----------------
// 16 lanes = 16 states of one (b,k,d) chain; lane 0 accumulates y via atomicAdd.
__global__ void k_scan(const float* __restrict__ Xdbl, const float* __restrict__ Xconv,
                       const float* __restrict__ A_logs, const float* __restrict__ dtw,
                       const float* __restrict__ dtb, const float* __restrict__ Dsp,
                       float* __restrict__ Y) {
  int idx = blockIdx.x * blockDim.x + threadIdx.x;
  int n = idx & 15;
  int scan = idx >> 4;                  // 0..1535
  int d = scan % DINN;
  int k = (scan / DINN) & (KDIR - 1);
  int b = scan / (DINN * KDIR);
  int kd = k * DINN + d;
  float Aval = -expf(A_logs[kd * DST + n]);
  float bias = dtb[kd];
  float w0 = dtw[kd * DTR + 0], w1 = dtw[kd * DTR + 1], w2 = dtw[kd * DTR + 2];
  float w3 = dtw[kd * DTR + 3], w4 = dtw[kd * DTR + 4], w5 = dtw[kd * DTR + 5];
  float Dp = Dsp[kd];
  const float* xd = Xdbl + (size_t)(b * KDIR + k) * Lsz * XDW;
  const float* xc = Xconv + (size_t)(b * DINN + d) * Lsz;
  float h = 0.f;
  for (int l = 0; l < Lsz; ++l) {
    const float* row = xd + (size_t)l * XDW;
    float raw = row[0] * w0 + row[1] * w1 + row[2] * w2 +
                row[3] * w3 + row[4] * w4 + row[5] * w5 + bias;
    float dt = (raw > 20.f) ? raw : log1pf(expf(raw));
    int pos = scan_pos(k, l);
    float u = xc[pos];
    float Bn = row[DTR + n];
    float Cn = row[DTR + DST + n];
    h = expf(dt * Aval) * h + dt * Bn * u;
    float yv = h * Cn;
    yv += __shfl_xor(yv, 1, 16);
    yv += __shfl_xor(yv, 2, 16);
    yv += __shfl_xor(yv, 4, 16);
    yv += __shfl_xor(yv, 8, 16);
    if (n == 0)
      atomicAdd(&Y[((size_t)b * Lsz + pos) * DINN + d], yv + u * Dp);
  }
}

// ---------------- launch ----------------
extern "C" void kernel_launch(void* const* d_in, const int* in_sizes, int n_in,
                              void* d_out, int out_size, void* d_ws, size_t ws_size,
                              hipStream_t stream) {
  (void)in_sizes; (void)n_in; (void)out_size; (void)ws_size;
  const float* x         = (const float*)d_in[0];
  const float* dw_w      = (const float*)d_in[1];
  const float* dw_b      = (const float*)d_in[2];
  const float* bn1_g     = (const float*)d_in[3];
  const float* bn1_b     = (const float*)d_in[4];
  const float* bn1_m     = (const float*)d_in[5];
  const float* bn1_v     = (const float*)d_in[6];
  const float* pw_w      = (const float*)d_in[7];
  const float* pw_b      = (const float*)d_in[8];
  const float* bn2_g     = (const float*)d_in[9];
  const float* bn2_b     = (const float*)d_in[10];
  const float* bn2_m     = (const float*)d_in[11];
  const float* bn2_v     = (const float*)d_in[12];
  const float* ln_g      = (const float*)d_in[13];
  const float* ln_b      = (const float*)d_in[14];
  const float* in_proj_w = (const float*)d_in[15];
  const float* conv_w    = (const float*)d_in[16];
  const float* conv_b    = (const float*)d_in[17];
  const float* x_proj_w  = (const float*)d_in[18];
  const float* dt_w      = (const float*)d_in[19];
  const float* dt_b      = (const float*)d_in[20];
  const float* A_logs    = (const float*)d_in[21];
  const float* Ds        = (const float*)d_in[22];
  const float* onorm_g   = (const float*)d_in[23];
  const float* onorm_b   = (const float*)d_in[24];
  const float* out_proj_w= (const float*)d_in[25];
  const float* fus_w     = (const float*)d_in[26];
  const float* fus_b     = (const float*)d_in[27];
  const float* fln_g     = (const float*)d_in[28];
  const float* fln_b     = (const float*)d_in[29];
  float* out = (float*)d_out;

  float* ws = (float*)d_ws;
  float* xn        = ws; ws += NROW * DIMc;
  float* xz        = ws; ws += NROW * 2 * DINN;
  float* xconv     = ws; ws += Bsz * DINN * Lsz;
  float* xdbl      = ws; ws += (size_t)Bsz * KDIR * Lsz * XDW;
  float* yb        = ws; ws += NROW * DINN;
  float* yg        = ws; ws += NROW * DINN;
  float* cbr       = ws; ws += NROW * DIMc;
  float* conv_out  = ws; ws += NROW * DIMc;
  float* mamba_out = ws; ws += NROW * DIMc;
  float* fused     = ws; ws += NROW * DIMc;

  // 1. LN(x) -> xn
  k_ln<<<dim3((NROW + 255) / 256), dim3(256), 0, stream>>>(x, ln_g, ln_b, xn, NROW, DIMc);
  // 2. in_proj GEMM (WMMA f32)
  k_gemm_inproj<<<dim3(NROW / 16, (2 * DINN) / 16), dim3(32), 0, stream>>>(xn, in_proj_w, xz);
  // 3. mamba dwconv + silu -> xconv (B,192,L)
  {
    int nt = Bsz * DINN * Lsz;
    k_conv_mamba<<<dim3((nt + 255) / 256), dim3(256), 0, stream>>>(xz, conv_w, conv_b, xconv);
  }
  // 4. x_proj GEMM per (b,k) direction (WMMA f32)
  k_gemm_xproj<<<dim3(Lsz / 16, XDW / 16, Bsz * KDIR), dim3(32), 0, stream>>>(xconv, x_proj_w, xdbl);
  // 5. zero y accumulator
  {
    int ny = NROW * DINN;
    k_zero<<<dim3((ny + 255) / 256), dim3(256), 0, stream>>>(yb, ny);
  }
  // 6. selective scan (4 directions fused via atomic accumulate)
  k_scan<<<dim3((Bsz * KDIR * DINN * DST) / 256), dim3(256), 0, stream>>>(
      xdbl, xconv, A_logs, dt_w, dt_b, Ds, yb);
  // 7. out_norm LN + silu(z) gate -> yg
  k_gate<<<dim3((NROW + 255) / 256), dim3(256), 0, stream>>>(yb, xz, onorm_g, onorm_b, yg);
  // 8. out_proj GEMM (WMMA f32)
  k_gemm_outproj<<<dim3(NROW / 16, DIMc / 16), dim3(32), 0, stream>>>(yg, out_proj_w, mamba_out);
  // 9. conv branch dw + bn1 + relu -> cbr
  {
    int nt = NROW * DIMc;
    k_conv_branch<<<dim3((nt + 255) / 256), dim3(256), 0, stream>>>(
        x, dw_w, dw_b, bn1_g, bn1_b, bn1_m, bn1_v, cbr);
  }
  // 10. pointwise conv GEMM + bn2 (WMMA f32)
  k_gemm_pw<<<dim3(NROW / 16, DIMc / 16), dim3(32), 0, stream>>>(
      cbr, pw_w, pw_b, bn2_g, bn2_b, bn2_m, bn2_v, conv_out);
  // 11. fusion GEMM over concat (WMMA f32)
  k_gemm_fus<<<dim3(NROW / 16, DIMc / 16), dim3(32), 0, stream>>>(
      conv_out, mamba_out, fus_w, fus_b, fused);
  // 12. final LN + residual
  k_final<<<dim3((NROW + 255) / 256), dim3(256), 0, stream>>>(x, fused, fln_g, fln_b, out);
}